// SPINN_18949395710773
// MI455X (gfx1250) — compile-verified
//
#include <hip/hip_runtime.h>
#include <hip/hip_bf16.h>

// ---------------------------------------------------------------------------
// SPINN on MI455X (gfx1250):
//   Phase 1: word projection GEMM  [16384,1024]@[1024,1024]+b  (fp32 WMMA)
//   Phase 2: 511 sequential TreeLSTM steps; each step =
//            g = stack_h[li]@red_lW + red_lb + stack_h[ri]@red_rW  (fp32 WMMA,
//            A rows gathered indirectly via stack pointers), then elementwise
//            gates + stack update.
// Matrix ops: V_WMMA_F32_16X16X4_F32.
// Staging:    GLOBAL_LOAD_ASYNC_TO_LDS_B128 (ASYNCcnt) + double-buffered LDS.
// ---------------------------------------------------------------------------

typedef __attribute__((ext_vector_type(2))) float v2f;
typedef __attribute__((ext_vector_type(8))) float v8f;
typedef __attribute__((ext_vector_type(4))) float f32x4;
typedef int v4i_gnu __attribute__((vector_size(16)));   // builtin's 16B unit

#define BATCH   64
#define SEQL    256
#define EMBED   1024
#define HSIZE   512
#define TSTEPS  511          // 2*L - 1
#define SDEPTH  257          // L + 1
#define GWIDTH  2560         // 5*HSIZE

#define AS1 __attribute__((address_space(1)))
#define AS3 __attribute__((address_space(3)))

#if __has_builtin(__builtin_amdgcn_global_load_async_to_lds_b128) && \
    __has_builtin(__builtin_amdgcn_s_wait_asynccnt)
#define HAVE_ASYNC_LDS 1
#endif

__device__ __forceinline__ int iclamp(int x, int lo, int hi) {
    return x < lo ? lo : (x > hi ? hi : x);
}
__device__ __forceinline__ float sigmf(float x) {
    return 1.0f / (1.0f + __expf(-x));
}

// 16B global -> LDS copy (async on gfx1250: no VGPR round trip, ASYNCcnt).
__device__ __forceinline__ void cp16_g2l(const float* __restrict__ g,
                                         float* __restrict__ l) {
#ifdef HAVE_ASYNC_LDS
    __builtin_amdgcn_global_load_async_to_lds_b128(
        (AS1 v4i_gnu*)g, (AS3 v4i_gnu*)l, /*imm offset*/0, /*cpol*/0);
#else
    *(f32x4*)l = *(const f32x4*)g;
#endif
}
__device__ __forceinline__ void cp_join() {
#ifdef HAVE_ASYNC_LDS
    __builtin_amdgcn_s_wait_asynccnt(0);
#endif
}

// ---------------------------------------------------------------------------
// Kernel 1: word projection.  C[M,N] = A[M,K] @ W[K,N] + bias[N]
// M=16384, N=1024, K=1024.  Block = 256 thr (8 waves), tile 128(M) x 128(N).
// Wave w owns rows [w*16, w*16+16) of the block tile, 8 n-tiles of 16.
// Double-buffered LDS, async staging overlapped with WMMA compute.
// ---------------------------------------------------------------------------
__global__ __launch_bounds__(256) void spinn_wordproj_kernel(
    const float* __restrict__ A, const float* __restrict__ W,
    const float* __restrict__ bias, float* __restrict__ C,
    int M, int N, int K) {
    __shared__ __align__(16) float As[2][128][36];   // padded stride vs banks
    __shared__ __align__(16) float Ws[2][32][132];

    const int tid  = threadIdx.x;
    const int wave = tid >> 5;
    const int lane = tid & 31;
    const int m15  = lane & 15;
    const int kh   = lane >> 4;          // 0: K pair {0,1}, 1: K pair {2,3}
    const int m0   = blockIdx.x * 128;
    const int n0   = blockIdx.y * 128;

    v8f acc[8] = {};

    auto stage = [&](int buf, int ks) {
        // A tile 128x32: 1024 x 16B, 4 per thread
        #pragma unroll
        for (int i = 0; i < 4; ++i) {
            int slot = tid + i * 256;          // 0..1023
            int r  = slot >> 3;                // 8 x 16B per row
            int cv = slot & 7;
            cp16_g2l(A + (size_t)(m0 + r) * K + ks + cv * 4, &As[buf][r][cv * 4]);
        }
        // W tile 32x128: 1024 x 16B, 4 per thread
        #pragma unroll
        for (int i = 0; i < 4; ++i) {
            int slot = tid + i * 256;
            int r  = slot >> 5;                // 32 x 16B per row
            int cv = slot & 31;
            cp16_g2l(W + (size_t)(ks + r) * N + n0 + cv * 4, &Ws[buf][r][cv * 4]);
        }
    };

    stage(0, 0);
    cp_join();
    __syncthreads();

    int buf = 0;
    for (int ks = 0; ks < K; ks += 32) {
        if (ks + 32 < K) stage(buf ^ 1, ks + 32);   // overlap next slice

        #pragma unroll
        for (int kk = 0; kk < 32; kk += 4) {
            v2f a;
            a.x = As[buf][wave * 16 + m15][kk + kh * 2 + 0];
            a.y = As[buf][wave * 16 + m15][kk + kh * 2 + 1];
            #pragma unroll
            for (int t = 0; t < 8; ++t) {
                v2f b;
                b.x = Ws[buf][kk + kh * 2 + 0][t * 16 + m15];
                b.y = Ws[buf][kk + kh * 2 + 1][t * 16 + m15];
                acc[t] = __builtin_amdgcn_wmma_f32_16x16x4_f32(
                    false, a, false, b, (short)0, acc[t], false, false);
            }
        }
        cp_join();        // next slice landed in LDS
        __syncthreads();  // all waves done reading + all data visible
        buf ^= 1;
    }

    // store with bias.  C/D layout: VGPR i -> row (i + 8*kh), col = lane&15
    #pragma unroll
    for (int t = 0; t < 8; ++t) {
        int col = n0 + t * 16 + m15;
        float bv = bias[col];
        #pragma unroll
        for (int i = 0; i < 8; ++i) {
            int row = m0 + wave * 16 + i + 8 * kh;
            C[(size_t)row * N + col] = acc[t][i] + bv;
        }
    }
}

// ---------------------------------------------------------------------------
// Kernel 2 (per step): g[64,2560] = stack_h[b,li]@red_lW + red_lb
//                                  + stack_h[b,ri]@red_rW
// A rows gathered indirectly through ptr[] (li = ptr-2, ri = ptr-1, clamped).
// Block = 128 thr (4 waves), tile 64(M=batch) x 128(N); grid.x = 2560/128 = 20.
// ---------------------------------------------------------------------------
__global__ __launch_bounds__(128) void spinn_reduce_gemm_kernel(
    const float* __restrict__ stack_h, const int* __restrict__ ptr,
    const float* __restrict__ lW, const float* __restrict__ rW,
    const float* __restrict__ lb, float* __restrict__ g) {
    __shared__ __align__(16) float As[2][64][36];
    __shared__ __align__(16) float Ws[2][32][132];
    __shared__ int rowoff[64];

    const int tid  = threadIdx.x;
    const int wave = tid >> 5;           // 0..3
    const int lane = tid & 31;
    const int m15  = lane & 15;
    const int kh   = lane >> 4;
    const int n0   = blockIdx.x * 128;

    v8f acc[8] = {};

    for (int phase = 0; phase < 2; ++phase) {
        const float* W = phase ? rW : lW;

        if (tid < BATCH) {
            int p   = ptr[tid];
            int idx = iclamp(phase ? (p - 1) : (p - 2), 0, SDEPTH - 1);
            rowoff[tid] = (tid * SDEPTH + idx) * HSIZE;
        }
        __syncthreads();

        auto stage = [&](int buf, int ks) {
            // gather A tile 64x32 from stack rows: 512 x 16B, 4 per thread
            #pragma unroll
            for (int i = 0; i < 4; ++i) {
                int slot = tid + i * 128;      // 0..511
                int r  = slot >> 3;
                int cv = slot & 7;
                cp16_g2l(stack_h + rowoff[r] + ks + cv * 4, &As[buf][r][cv * 4]);
            }
            // W tile 32x128: 1024 x 16B, 8 per thread
            #pragma unroll
            for (int i = 0; i < 8; ++i) {
                int slot = tid + i * 128;
                int r  = slot >> 5;
                int cv = slot & 31;
                cp16_g2l(W + (size_t)(ks + r) * GWIDTH + n0 + cv * 4,
                         &Ws[buf][r][cv * 4]);
            }
        };

        stage(0, 0);
        cp_join();
        __syncthreads();

        int buf = 0;
        for (int ks = 0; ks < HSIZE; ks += 32) {
            if (ks + 32 < HSIZE) stage(buf ^ 1, ks + 32);

            #pragma unroll
            for (int kk = 0; kk < 32; kk += 4) {
                v2f a;
                a.x = As[buf][wave * 16 + m15][kk + kh * 2 + 0];
                a.y = As[buf][wave * 16 + m15][kk + kh * 2 + 1];
                #pragma unroll
                for (int t = 0; t < 8; ++t) {
                    v2f b;
                    b.x = Ws[buf][kk + kh * 2 + 0][t * 16 + m15];
                    b.y = Ws[buf][kk + kh * 2 + 1][t * 16 + m15];
                    acc[t] = __builtin_amdgcn_wmma_f32_16x16x4_f32(
                        false, a, false, b, (short)0, acc[t], false, false);
                }
            }
            cp_join();
            __syncthreads();
            buf ^= 1;
        }
    }

    #pragma unroll
    for (int t = 0; t < 8; ++t) {
        int col = n0 + t * 16 + m15;
        float bv = lb[col];
        #pragma unroll
        for (int i = 0; i < 8; ++i) {
            int row = wave * 16 + i + 8 * kh;   // row == batch index
            g[(size_t)row * GWIDTH + col] = acc[t][i] + bv;
        }
    }
}

// ---------------------------------------------------------------------------
// Kernel 3 (per step): TreeLSTM gates + stack update.  One block per batch row.
// ---------------------------------------------------------------------------
__global__ __launch_bounds__(256) void spinn_tree_step_kernel(
    const float* __restrict__ g, const float* __restrict__ proj,
    const int* __restrict__ trans, float* __restrict__ stack_h,
    float* __restrict__ stack_c, int* __restrict__ ptr,
    int* __restrict__ bptr, int t) {
    const int b   = blockIdx.x;
    const int act = trans[b * TSTEPS + t];
    const bool is_shift = (act == 3);
    const bool is_red   = (act == 2);
    const bool active   = is_shift || is_red;

    const int p  = ptr[b];
    const int bp = bptr[b];
    const int ri = iclamp(p - 1, 0, SDEPTH - 1);
    const int li = iclamp(p - 2, 0, SDEPTH - 1);
    const int widx = iclamp(SEQL - 1 - bp, 0, SEQL - 1);
    const int wpos = iclamp(is_shift ? p : p - 2, 0, SDEPTH - 1);

    if (active) {
        const float* grow = g + (size_t)b * GWIDTH;
        const float* prow = proj + ((size_t)b * SEQL + widx) * (2 * HSIZE);
        for (int n = threadIdx.x; n < HSIZE; n += blockDim.x) {
            float av = grow[n];
            float iv = grow[HSIZE + n];
            float f1 = grow[2 * HSIZE + n];
            float f2 = grow[3 * HSIZE + n];
            float ov = grow[4 * HSIZE + n];
            float lc = stack_c[((size_t)b * SDEPTH + li) * HSIZE + n];
            float rc = stack_c[((size_t)b * SDEPTH + ri) * HSIZE + n];
            float c_red = tanhf(av) * sigmf(iv) + sigmf(f1) * lc + sigmf(f2) * rc;
            float h_red = sigmf(ov) * tanhf(c_red);
            float wh = prow[n];
            float wc = prow[HSIZE + n];
            float nh = is_shift ? wh : h_red;
            float nc = is_shift ? wc : c_red;
            stack_h[((size_t)b * SDEPTH + wpos) * HSIZE + n] = nh;
            stack_c[((size_t)b * SDEPTH + wpos) * HSIZE + n] = nc;
        }
    }
    __syncthreads();
    if (threadIdx.x == 0) {
        ptr[b]  = p + (is_shift ? 1 : 0) - (is_red ? 1 : 0);
        bptr[b] = bp + (is_shift ? 1 : 0);
    }
}

// ---------------------------------------------------------------------------
// Init (zero mutable state each call) and finalize (copy stack top to out).
// ---------------------------------------------------------------------------
__global__ void spinn_init_kernel(float* __restrict__ stack_h,
                                  float* __restrict__ stack_c,
                                  int* __restrict__ ptr, int* __restrict__ bptr,
                                  size_t nstack) {
    size_t i = (size_t)blockIdx.x * blockDim.x + threadIdx.x;
    size_t stride = (size_t)gridDim.x * blockDim.x;
    for (size_t j = i; j < nstack; j += stride) {
        stack_h[j] = 0.0f;
        stack_c[j] = 0.0f;
    }
    if (i < BATCH) { ptr[i] = 0; bptr[i] = 0; }
}

__global__ void spinn_finalize_kernel(const float* __restrict__ stack_h,
                                      const int* __restrict__ ptr,
                                      float* __restrict__ out) {
    int b = blockIdx.x;
    int top = iclamp(ptr[b] - 1, 0, SDEPTH - 1);
    for (int n = threadIdx.x; n < HSIZE; n += blockDim.x)
        out[(size_t)b * HSIZE + n] = stack_h[((size_t)b * SDEPTH + top) * HSIZE + n];
}

// ---------------------------------------------------------------------------
extern "C" void kernel_launch(void* const* d_in, const int* in_sizes, int n_in,
                              void* d_out, int out_size, void* d_ws, size_t ws_size,
                              hipStream_t stream) {
    const float* sentence    = (const float*)d_in[0];
    const int*   transitions = (const int*)d_in[1];
    const float* word_W      = (const float*)d_in[2];
    const float* word_b      = (const float*)d_in[3];
    const float* red_lW      = (const float*)d_in[4];
    const float* red_lb      = (const float*)d_in[5];
    const float* red_rW      = (const float*)d_in[6];

    // workspace carve (floats)
    float* proj    = (float*)d_ws;                              // [16384, 1024]
    float* stack_h = proj + (size_t)BATCH * SEQL * 2 * HSIZE;   // [64, 257, 512]
    float* stack_c = stack_h + (size_t)BATCH * SDEPTH * HSIZE;
    float* gbuf    = stack_c + (size_t)BATCH * SDEPTH * HSIZE;  // [64, 2560]
    int*   ptr     = (int*)(gbuf + (size_t)BATCH * GWIDTH);
    int*   bptr    = ptr + BATCH;

    const size_t nstack = (size_t)BATCH * SDEPTH * HSIZE;
    spinn_init_kernel<<<1024, 256, 0, stream>>>(stack_h, stack_c, ptr, bptr, nstack);

    // word projection: M=16384, N=1024, K=1024
    spinn_wordproj_kernel<<<dim3(128, 8), 256, 0, stream>>>(
        sentence, word_W, word_b, proj, BATCH * SEQL, 2 * HSIZE, EMBED);

    // sequential shift-reduce scan
    for (int t = 0; t < TSTEPS; ++t) {
        spinn_reduce_gemm_kernel<<<GWIDTH / 128, 128, 0, stream>>>(
            stack_h, ptr, red_lW, red_rW, red_lb, gbuf);
        spinn_tree_step_kernel<<<BATCH, 256, 0, stream>>>(
            gbuf, proj, transitions, stack_h, stack_c, ptr, bptr, t);
    }

    spinn_finalize_kernel<<<BATCH, 256, 0, stream>>>(stack_h, ptr, (float*)d_out);
}